// OpenVLALMDecodeAttention_10101763080524
// MI455X (gfx1250) — compile-verified
//
#include <hip/hip_runtime.h>

typedef __attribute__((ext_vector_type(2))) float v2f;
typedef __attribute__((ext_vector_type(8))) float v8f;

#define HIDDEN    4096
#define N_HEADS   32
#define HEAD_DIM  128
#define CACHE_LEN 512
#define BATCH     32

// Y[M=32][N] = A[32][K] @ W[N][K]^T  with f32 WMMA 16x16x4.
// Block = 256 threads = 8 waves. One 16-wide N tile per block; the 8 waves
// split K eight ways (kslice = K/8) and reduce partial C tiles through LDS.
// Each wave double-buffers 16-K chunks (12 b64 loads in flight while the
// previous chunk's 8 WMMAs issue) so a single wave overlaps HBM and matrix ops.
// blockIdx.y selects between two independent (W, Y) pairs so q and v
// projections run concurrently in one launch.
__global__ __launch_bounds__(256) void gemm_awt_wmma(const float* __restrict__ A,
                                                     const float* __restrict__ W0,
                                                     float* __restrict__ Y0,
                                                     const float* __restrict__ W1,
                                                     float* __restrict__ Y1,
                                                     int K, int N) {
    const float* __restrict__ W = blockIdx.y ? W1 : W0;
    float* __restrict__       Y = blockIdx.y ? Y1 : Y0;

    const int lane = threadIdx.x & 31;
    const int wave = threadIdx.x >> 5;   // 0..7 : K slice
    const int n0   = blockIdx.x * 16;
    const int lr   = lane & 15;          // row (A) / col (B) inside tile
    const int kh   = (lane >> 4) * 2;    // lanes 0-15: K=0,1 ; lanes 16-31: K=2,3

    const int kslice = K >> 3;           // 512
    const int kbeg   = wave * kslice;

    const float* arow0 = A + (size_t)(lr)      * K + kbeg + kh;
    const float* arow1 = A + (size_t)(lr + 16) * K + kbeg + kh;
    const float* wrow  = W + (size_t)(n0 + lr) * K + kbeg + kh;

    v8f c0 = {};
    v8f c1 = {};

    // current / next 16-K chunks (4 WMMA steps each)
    v2f ca0[4], ca1[4], cb[4];
#pragma unroll
    for (int u = 0; u < 4; ++u) {
        ca0[u] = *(const v2f*)(arow0 + 4 * u);
        ca1[u] = *(const v2f*)(arow1 + 4 * u);
        cb[u]  = *(const v2f*)(wrow  + 4 * u);
    }

    for (int k = 16; k < kslice; k += 16) {
        v2f na0[4], na1[4], nb[4];
#pragma unroll
        for (int u = 0; u < 4; ++u) {       // prefetch next chunk
            na0[u] = *(const v2f*)(arow0 + k + 4 * u);
            na1[u] = *(const v2f*)(arow1 + k + 4 * u);
            nb[u]  = *(const v2f*)(wrow  + k + 4 * u);
        }
#pragma unroll
        for (int u = 0; u < 4; ++u) {       // consume current chunk
            c0 = __builtin_amdgcn_wmma_f32_16x16x4_f32(false, ca0[u], false, cb[u],
                                                       (short)0, c0, false, false);
            c1 = __builtin_amdgcn_wmma_f32_16x16x4_f32(false, ca1[u], false, cb[u],
                                                       (short)0, c1, false, false);
        }
#pragma unroll
        for (int u = 0; u < 4; ++u) {
            ca0[u] = na0[u]; ca1[u] = na1[u]; cb[u] = nb[u];
        }
    }
#pragma unroll
    for (int u = 0; u < 4; ++u) {           // epilogue chunk
        c0 = __builtin_amdgcn_wmma_f32_16x16x4_f32(false, ca0[u], false, cb[u],
                                                   (short)0, c0, false, false);
        c1 = __builtin_amdgcn_wmma_f32_16x16x4_f32(false, ca1[u], false, cb[u],
                                                   (short)0, c1, false, false);
    }

    // K-slice reduction across the 8 waves through LDS.
    // C/D layout: VGPR r -> M = r (lanes 0-15) / r+8 (lanes 16-31), N = lane&15.
    __shared__ float redbuf[8 * 512];
    float* myb = redbuf + wave * 512;
    const int rbase = (lane >> 4) * 8;
#pragma unroll
    for (int r = 0; r < 8; ++r) {
        myb[(rbase + r) * 16 + lr]       = c0[r];   // rows 0..15
        myb[256 + (rbase + r) * 16 + lr] = c1[r];   // rows 16..31
    }
    __syncthreads();

    const int tid = threadIdx.x;
#pragma unroll
    for (int e = tid; e < 512; e += 256) {
        float s = 0.0f;
#pragma unroll
        for (int w = 0; w < 8; ++w) s += redbuf[w * 512 + e];
        const int row = e >> 4;          // 0..31
        const int col = n0 + (e & 15);
        Y[(size_t)row * N + col] = s;
    }
}

// One block (8 wave32) per (batch, head). softmax(q.K^T/sqrt(d)).V + v_new.
// (softmax over the single new key is identically 1 -> new_out == v_new,
//  so k_new / wk are dead and never computed.)
__global__ __launch_bounds__(256) void attn_decode(const float* __restrict__ q_all,
                                                   const float* __restrict__ v_new_all,
                                                   const float* __restrict__ k_cache,
                                                   const float* __restrict__ v_cache,
                                                   float* __restrict__ attn_out) {
    __shared__ __align__(16) float q_s[HEAD_DIM];
    __shared__ float sc[CACHE_LEN];
    __shared__ float red[256];
    __shared__ __align__(8) float pout[4][HEAD_DIM];

    const int    bh      = blockIdx.x;                 // b*N_HEADS + h
    const int    tid     = threadIdx.x;
    const int    lane    = tid & 31;
    const int    wave    = tid >> 5;
    const size_t kv_base = (size_t)bh * CACHE_LEN * HEAD_DIM;
    const size_t q_base  = (size_t)bh * HEAD_DIM;      // == b*4096 + h*128

    if (tid < HEAD_DIM) q_s[tid] = q_all[q_base + tid];
    __syncthreads();

    const float4 qv    = *reinterpret_cast<const float4*>(&q_s[lane * 4]);
    const float  scale = 0.08838834764831845f;         // 1/sqrt(128)

    // scores: one key per wave per step; float4 loads = 512B coalesced per row
    for (int kk = wave; kk < CACHE_LEN; kk += 8) {
        const float4 kvv = *reinterpret_cast<const float4*>(
            k_cache + kv_base + (size_t)kk * HEAD_DIM + lane * 4);
        float p = qv.x * kvv.x + qv.y * kvv.y + qv.z * kvv.z + qv.w * kvv.w;
        p += __shfl_xor(p, 16);
        p += __shfl_xor(p, 8);
        p += __shfl_xor(p, 4);
        p += __shfl_xor(p, 2);
        p += __shfl_xor(p, 1);
        if (lane == 0) sc[kk] = p * scale;
    }
    __syncthreads();

    // softmax over 512 scores (keep exps unnormalized, fold 1/sum at the end)
    red[tid] = fmaxf(sc[tid], sc[tid + 256]);
    __syncthreads();
    for (int s = 128; s > 0; s >>= 1) {
        if (tid < s) red[tid] = fmaxf(red[tid], red[tid + s]);
        __syncthreads();
    }
    const float mx = red[0];
    __syncthreads();
    const float e0 = __expf(sc[tid] - mx);
    const float e1 = __expf(sc[tid + 256] - mx);
    sc[tid] = e0;
    sc[tid + 256] = e1;
    red[tid] = e0 + e1;
    __syncthreads();
    for (int s = 128; s > 0; s >>= 1) {
        if (tid < s) red[tid] += red[tid + s];
        __syncthreads();
    }
    const float inv_sum = 1.0f / red[0];
    __syncthreads();

    // out[d] = inv_sum * sum_k sc[k]*V[k][d] + v_new[d]
    // 4 key-groups x 64 threads; each thread owns a d-pair -> float2 loads,
    // 64 lanes x 8B = 512B coalesced per V row.
    const int    grp  = tid >> 6;         // 0..3 -> keys grp*128 .. grp*128+127
    const int    dt   = tid & 63;         // d pair index
    const float* vptr = v_cache + kv_base + (size_t)grp * 128 * HEAD_DIM + dt * 2;
    float ax = 0.0f, ay = 0.0f;
#pragma unroll 4
    for (int kk = 0; kk < 128; ++kk) {
        const float2 vv = *reinterpret_cast<const float2*>(vptr + (size_t)kk * HEAD_DIM);
        const float  p  = sc[grp * 128 + kk];
        ax += p * vv.x;
        ay += p * vv.y;
    }
    pout[grp][dt * 2]     = ax;
    pout[grp][dt * 2 + 1] = ay;
    __syncthreads();
    if (tid < 64) {
        const int d0 = tid * 2;
        const float s0 = pout[0][d0] + pout[1][d0] + pout[2][d0] + pout[3][d0];
        const float s1 = pout[0][d0 + 1] + pout[1][d0 + 1] + pout[2][d0 + 1] + pout[3][d0 + 1];
        attn_out[q_base + d0]     = s0 * inv_sum + v_new_all[q_base + d0];
        attn_out[q_base + d0 + 1] = s1 * inv_sum + v_new_all[q_base + d0 + 1];
    }
}

extern "C" void kernel_launch(void* const* d_in, const int* in_sizes, int n_in,
                              void* d_out, int out_size, void* d_ws, size_t ws_size,
                              hipStream_t stream) {
    const float* x       = (const float*)d_in[0];
    const float* k_cache = (const float*)d_in[1];
    const float* v_cache = (const float*)d_in[2];
    const float* wq      = (const float*)d_in[3];
    // d_in[4] (wk) intentionally unused: softmax over one key == 1 exactly.
    const float* wv      = (const float*)d_in[5];
    const float* wo      = (const float*)d_in[6];
    float*       out     = (float*)d_out;

    float* q_ws    = (float*)d_ws;                 // 32*4096 f32
    float* v_ws    = q_ws + BATCH * HIDDEN;        // 32*4096 f32
    float* attn_ws = v_ws + BATCH * HIDDEN;        // 32*4096 f32

    const dim3 gblk(256);

    // fused q + v projections: gridDim.y = 2 selects (wq->q_ws) / (wv->v_ws)
    gemm_awt_wmma<<<dim3(HIDDEN / 16, 2), gblk, 0, stream>>>(
        x, wq, q_ws, wv, v_ws, HIDDEN, HIDDEN);

    attn_decode<<<BATCH * N_HEADS, 256, 0, stream>>>(
        q_ws, v_ws, k_cache, v_cache, attn_ws);

    // output projection
    gemm_awt_wmma<<<dim3(HIDDEN / 16, 1), gblk, 0, stream>>>(
        attn_ws, wo, out, wo, out, HIDDEN, HIDDEN);
}